// SpatialDecoder_26637387170114
// MI455X (gfx1250) — compile-verified
//
#include <hip/hip_runtime.h>
#include <hip/hip_bf16.h>

// ---------------------------------------------------------------------------
// GAT (3-layer, D=128) for gfx1250.
//  - GEMM h = x @ W via V_WMMA_F32_16X16X4_F32 (native fp32 WMMA).
//  - Edge softmax-aggregation via global f32 atomics (L2-resident targets).
// ---------------------------------------------------------------------------

typedef float v2f __attribute__((ext_vector_type(2)));
typedef float v8f __attribute__((ext_vector_type(8)));

#define D 128
#define LDS_STRIDE 132   // 128 + 4 pad to break bank conflicts

// ---------------------------------------------------------------------------
// GEMM: H[N,128] = X[N,128] @ W[128,128], fp32 WMMA 16x16x4.
// Block = 256 threads (8 waves). Tile = 32 rows x 128 cols.
// Wave w: mtile = w&1 (16 rows), ntile = w>>1 (cols nt*16 and nt*16+64).
// A-operand layout (ISA 7.12.2, 32-bit A 16x4):
//   lanes 0-15: M=lane, VGPR0=K0, VGPR1=K0+1 ; lanes 16-31: K0+2, K0+3.
// B-operand (4x16): lanes 0-15 N=lane holding K rows {k,k+1}; lanes16-31 {k+2,k+3}.
// C/D (16x16 f32): VGPR r -> M=r (lanes0-15), M=r+8 (lanes16-31), N=lane&15.
// ---------------------------------------------------------------------------
__global__ __launch_bounds__(256)
void gat_gemm_wmma(const float* __restrict__ X, const float* __restrict__ W,
                   float* __restrict__ H, int N)
{
    __shared__ float xs[32][LDS_STRIDE];

    const int tid  = threadIdx.x;
    const int lane = tid & 31;
    const int w    = tid >> 5;
    const int rowbase = blockIdx.x * 32;

    // Cooperative load of the 32x128 x-tile into LDS (float4 per thread x4).
    for (int t = 0; t < 4; ++t) {
        int idx = tid + t * 256;          // 0..1023 float4 slots
        int row = idx >> 5;               // 32 float4 per row
        int c4  = (idx & 31) << 2;        // column (floats)
        int gr  = rowbase + row;
        float4 v = make_float4(0.f, 0.f, 0.f, 0.f);
        if (gr < N) v = *(const float4*)&X[(size_t)gr * D + c4];
        *(float4*)&xs[row][c4] = v;
    }
    __syncthreads();

    const int mtile = w & 1;
    const int nt    = w >> 1;
    const int l15   = lane & 15;
    const int lhi   = lane >> 4;          // 0 or 1
    const int arow  = mtile * 16 + l15;
    const int koff  = lhi * 2;
    const int c0    = nt * 16 + l15;
    const int c1    = c0 + 64;

    v8f acc0 = {};
    v8f acc1 = {};

#pragma unroll
    for (int k0 = 0; k0 < D; k0 += 4) {
        const int kk = k0 + koff;
        v2f a = *(const v2f*)&xs[arow][kk];
        v2f b0, b1;
        b0.x = W[(size_t)kk * D + c0];
        b0.y = W[(size_t)(kk + 1) * D + c0];
        b1.x = W[(size_t)kk * D + c1];
        b1.y = W[(size_t)(kk + 1) * D + c1];
        acc0 = __builtin_amdgcn_wmma_f32_16x16x4_f32(false, a, false, b0,
                                                     (short)0, acc0, false, false);
        acc1 = __builtin_amdgcn_wmma_f32_16x16x4_f32(false, a, false, b1,
                                                     (short)0, acc1, false, false);
    }

#pragma unroll
    for (int r = 0; r < 8; ++r) {
        int grow = rowbase + mtile * 16 + lhi * 8 + r;
        if (grow < N) {
            H[(size_t)grow * D + c0] = acc0[r];
            H[(size_t)grow * D + c1] = acc1[r];
        }
    }
}

// ---------------------------------------------------------------------------
// Per-row dots: al_s[n] = h[n,:]·a_src, al_d[n] = h[n,:]·a_dst.
// One wave per row, 4 floats per lane, shuffle reduce.
// ---------------------------------------------------------------------------
__global__ __launch_bounds__(256)
void gat_rowdot(const float* __restrict__ H, const float* __restrict__ asrc,
                const float* __restrict__ adst, float* __restrict__ als,
                float* __restrict__ ald, int N)
{
    int wid  = (int)((blockIdx.x * blockDim.x + threadIdx.x) >> 5);
    int lane = threadIdx.x & 31;
    if (wid >= N) return;

    float4 h4 = *(const float4*)&H[(size_t)wid * D + lane * 4];
    float4 s4 = *(const float4*)&asrc[lane * 4];
    float4 d4 = *(const float4*)&adst[lane * 4];
    float ps = h4.x * s4.x + h4.y * s4.y + h4.z * s4.z + h4.w * s4.w;
    float pd = h4.x * d4.x + h4.y * d4.y + h4.z * d4.z + h4.w * d4.w;
#pragma unroll
    for (int off = 16; off > 0; off >>= 1) {
        ps += __shfl_down(ps, off, 32);
        pd += __shfl_down(pd, off, 32);
    }
    if (lane == 0) { als[wid] = ps; ald[wid] = pd; }
}

// ---------------------------------------------------------------------------
// Segment-softmax helpers.
// ---------------------------------------------------------------------------
__device__ inline void atomicMaxF32(float* addr, float val)
{
    if (val >= 0.0f) atomicMax((int*)addr, __float_as_int(val));
    else             atomicMin((unsigned int*)addr, __float_as_uint(val));
}

__global__ void gat_init_mz(float* __restrict__ m, float* __restrict__ z, int N)
{
    int i = blockIdx.x * blockDim.x + threadIdx.x;
    if (i < N) { m[i] = __int_as_float(0xff800000); z[i] = 0.0f; }
}

__global__ void gat_zero(float* __restrict__ p, size_t n)
{
    size_t i = (size_t)blockIdx.x * blockDim.x + threadIdx.x;
    if (i < n) p[i] = 0.0f;
}

// e = leaky_relu(al_s[src] + al_d[dst]); m[dst] = max(m[dst], e)
__global__ void gat_edge_logits(const int* __restrict__ src, const int* __restrict__ dst,
                                const float* __restrict__ als, const float* __restrict__ ald,
                                float* __restrict__ e, float* __restrict__ m,
                                int E, int Eall)
{
    int i = blockIdx.x * blockDim.x + threadIdx.x;
    if (i >= Eall) return;
    int s, d;
    if (i < E) { s = src[i]; d = dst[i]; } else { s = d = i - E; }
    float t = als[s] + ald[d];
    t = t > 0.0f ? t : 0.2f * t;
    e[i] = t;
    atomicMaxF32(&m[d], t);
}

// ex = exp(e - m[dst]); z[dst] += ex  (ex overwrites e)
__global__ void gat_edge_exp(const int* __restrict__ dst, float* __restrict__ e,
                             const float* __restrict__ m, float* __restrict__ z,
                             int E, int Eall)
{
    int i = blockIdx.x * blockDim.x + threadIdx.x;
    if (i >= Eall) return;
    int d = (i < E) ? dst[i] : i - E;
    float ex = __expf(e[i] - m[d]);
    e[i] = ex;
    atomicAdd(&z[d], ex);
}

// acc[dst,:] += (ex/z[dst]) * h[src,:].  One wave per edge, 4 feats/lane.
__global__ __launch_bounds__(256)
void gat_edge_scatter(const int* __restrict__ src, const int* __restrict__ dst,
                      const float* __restrict__ ex, const float* __restrict__ z,
                      const float* __restrict__ H, float* __restrict__ acc,
                      int E, int Eall)
{
    int wid  = (int)((blockIdx.x * blockDim.x + threadIdx.x) >> 5);
    int lane = threadIdx.x & 31;
    if (wid >= Eall) return;
    int s, d;
    if (wid < E) { s = src[wid]; d = dst[wid]; } else { s = d = wid - E; }
    float alpha = ex[wid] / z[d];
    float4 h4 = *(const float4*)&H[(size_t)s * D + lane * 4];
    float* ap = &acc[(size_t)d * D + lane * 4];
    atomicAdd(ap + 0, alpha * h4.x);
    atomicAdd(ap + 1, alpha * h4.y);
    atomicAdd(ap + 2, alpha * h4.z);
    atomicAdd(ap + 3, alpha * h4.w);
}

// out = relu(acc + b)
__global__ void gat_bias_relu(const float* __restrict__ acc, const float* __restrict__ b,
                              float* __restrict__ out, size_t n)
{
    size_t i = (size_t)blockIdx.x * blockDim.x + threadIdx.x;
    if (i >= n) return;
    float v = acc[i] + b[i & (D - 1)];
    out[i] = v > 0.0f ? v : 0.0f;
}

// ---------------------------------------------------------------------------
// Launcher.
// Inputs: x[N,128], edge_index[2,E], W[L,128,128], a_src[L,128], a_dst[L,128], b[L,128]
// ---------------------------------------------------------------------------
extern "C" void kernel_launch(void* const* d_in, const int* in_sizes, int n_in,
                              void* d_out, int out_size, void* d_ws, size_t ws_size,
                              hipStream_t stream)
{
    const float* x    = (const float*)d_in[0];
    const int*   eidx = (const int*)d_in[1];
    const float* Wl   = (const float*)d_in[2];
    const float* asl  = (const float*)d_in[3];
    const float* adl  = (const float*)d_in[4];
    const float* bl   = (const float*)d_in[5];

    const int N    = in_sizes[0] / D;
    const int E    = in_sizes[1] / 2;
    const int L    = in_sizes[2] / (D * D);
    const int Eall = E + N;
    const size_t ND = (size_t)N * D;

    const int* src = eidx;
    const int* dst = eidx + E;

    float* ws  = (float*)d_ws;
    float* wsH = ws;                 // [N,128] h buffer
    float* wsA = wsH + ND;           // [N,128] accumulator / ping-pong x
    float* als = wsA + ND;           // [N]
    float* ald = als + N;            // [N]
    float* mb  = ald + N;            // [N]
    float* zb  = mb + N;             // [N]
    float* eb  = zb + N;             // [Eall]
    float* outp = (float*)d_out;

    for (int l = 0; l < L; ++l) {
        const float* xin  = (l == 0) ? x : (l == 1 ? wsA : outp);
        float*       acc  = (l == 1) ? outp : wsA;
        float*       xout = (l == 2) ? outp : acc;
        const float* W    = Wl  + (size_t)l * D * D;
        const float* as_  = asl + (size_t)l * D;
        const float* ad_  = adl + (size_t)l * D;
        const float* b_   = bl  + (size_t)l * D;

        gat_gemm_wmma<<<(N + 31) / 32, 256, 0, stream>>>(xin, W, wsH, N);
        gat_rowdot<<<(N + 7) / 8, 256, 0, stream>>>(wsH, as_, ad_, als, ald, N);
        gat_init_mz<<<(N + 255) / 256, 256, 0, stream>>>(mb, zb, N);
        gat_zero<<<(unsigned)((ND + 255) / 256), 256, 0, stream>>>(acc, ND);
        gat_edge_logits<<<(Eall + 255) / 256, 256, 0, stream>>>(src, dst, als, ald,
                                                                eb, mb, E, Eall);
        gat_edge_exp<<<(Eall + 255) / 256, 256, 0, stream>>>(dst, eb, mb, zb, E, Eall);
        gat_edge_scatter<<<(Eall + 7) / 8, 256, 0, stream>>>(src, dst, eb, zb,
                                                             wsH, acc, E, Eall);
        gat_bias_relu<<<(unsigned)((ND + 255) / 256), 256, 0, stream>>>(acc, b_, xout, ND);
    }
}